// testAttention_37684043055338
// MI455X (gfx1250) — compile-verified
//
#include <hip/hip_runtime.h>
#include <hip/hip_bf16.h>

// ---------------------------------------------------------------------------
// CDNA5 (gfx1250, wave32) implementation of the Nystrom-like attention ref.
// All dense matmul work uses V_WMMA_F32_16X16X4_F32 (fp32 in, fp32 acc).
// ---------------------------------------------------------------------------

typedef __attribute__((ext_vector_type(2))) float v2f;
typedef __attribute__((ext_vector_type(8))) float v8f;

#define SEQ      3900
#define ROWS     7800      // 2 * 3900
#define MD       64
#define NB       12
#define NN       325
#define NC       6
#define NL       72        // NB * NC landmarks
#define NP       80        // padded to 5x 16-tiles
#define SCALE    0.35355339059327373f  // (64/8)^-0.5

__device__ __forceinline__ unsigned pcg_hash(unsigned v) {
    v = v * 747796405u + 2891336453u;
    unsigned w = ((v >> ((v >> 28u) + 4u)) ^ v) * 277803737u;
    return (w >> 22u) ^ w;
}

__device__ __forceinline__ float wave_max_f(float v) {
    #pragma unroll
    for (int m = 16; m > 0; m >>= 1) v = fmaxf(v, __shfl_xor(v, m, 32));
    return v;
}
__device__ __forceinline__ float wave_sum_f(float v) {
    #pragma unroll
    for (int m = 16; m > 0; m >>= 1) v += __shfl_xor(v, m, 32);
    return v;
}

// ---------------------------------------------------------------------------
// Kernel 1/4: C[M x N] = A[M x 64] @ W[N x 64]^T (+ bias), fp32 WMMA.
// One 16x16 output tile per wave; K=64 -> 16 v_wmma_f32_16x16x4_f32 steps.
// A layout (16x4 f32): lanes 0-15 M=lane, regs = K,K+1; lanes 16-31 K+2,K+3.
// B layout mirrors with N across lanes. C: reg r -> M = r + 8*(lane>=16).
// ---------------------------------------------------------------------------
__global__ void gemm_xwT_kernel(const float* __restrict__ A,
                                const float* __restrict__ W,
                                const float* __restrict__ bias,
                                float* __restrict__ C, int M, int N) {
    const int wid  = threadIdx.x >> 5;
    const int lane = threadIdx.x & 31;
    const int ntn  = N >> 4;
    const int mt   = (M + 15) >> 4;
    const int tile = blockIdx.x * (blockDim.x >> 5) + wid;
    if (tile >= mt * ntn) return;               // wave-uniform (EXEC all-1 for WMMA)
    const int tm = tile / ntn, tn = tile % ntn;
    const int l15 = lane & 15, hi = lane >> 4;
    const int mrow = tm * 16 + l15;
    const int mr   = mrow < M ? mrow : (M - 1); // clamp edge-tile loads
    const int ncol = tn * 16 + l15;
    const float* arow = A + (size_t)mr * 64;
    const float* wrow = W + (size_t)ncol * 64;
    v8f acc = {};
    #pragma unroll
    for (int s = 0; s < 16; ++s) {
        const int k0 = 4 * s + 2 * hi;
        v2f a = *(const v2f*)(arow + k0);
        v2f b = *(const v2f*)(wrow + k0);
        acc = __builtin_amdgcn_wmma_f32_16x16x4_f32(false, a, false, b,
                                                    (short)0, acc, false, false);
    }
    const int rbase = tm * 16 + (hi << 3);
    const float bb = bias ? bias[ncol] : 0.0f;
    #pragma unroll
    for (int r = 0; r < 8; ++r) {
        const int m = rbase + r;
        if (m < M) C[(size_t)m * N + ncol] = acc[r] + bb;
    }
}

// ---------------------------------------------------------------------------
// Kernel 2: per-(b,block,cluster,h) stats -> landmarks.
// grid = 24 (b,blk), block = 64 (h). Noise term: deterministic hash-based
// Box-Muller stand-in for jax.random.normal(key 42) (unverifiable here).
// ---------------------------------------------------------------------------
__global__ void landmarks_kernel(const float* __restrict__ qkv,
                                 const int* __restrict__ clusters,
                                 float* __restrict__ ql) {
    const int b   = blockIdx.x / NB;
    const int blk = blockIdx.x % NB;
    const int h   = threadIdx.x;
    float sums[NC], sqs[NC]; int cnts[NC];
    #pragma unroll
    for (int c = 0; c < NC; ++c) { sums[c] = 0.f; sqs[c] = 0.f; cnts[c] = 0; }
    const float* base = qkv + ((size_t)b * SEQ + (size_t)blk * NN) * 192;
    for (int n = 0; n < NN; ++n) {
        const int c = clusters[n];
        const float qv = base[(size_t)n * 192 + h] * SCALE;
        sums[c] += qv; sqs[c] += qv * qv; cnts[c] += 1;
    }
    #pragma unroll
    for (int c = 0; c < NC; ++c) {
        const float cnt  = (float)cnts[c];
        const float mean = sums[c] / cnt;
        const float var  = (sqs[c] - cnt * mean * mean) / (cnt - 1.0f);
        const float sd   = sqrtf(fmaxf(var, 0.0f));
        const unsigned seed =
            (unsigned)((((b * 64 + h) * NB + blk) * NC + c)) * 2654435761u + 0x9E3779B9u;
        float nm = 0.0f;
        #pragma unroll
        for (int s = 0; s < 10; ++s) {
            unsigned r0 = pcg_hash(seed + (unsigned)s * 0x85EBCA6Bu);
            unsigned r1 = pcg_hash(r0 ^ 0x68BC21EBu);
            float u1 = ((float)(r0 >> 8) + 1.0f) * (1.0f / 16777216.0f);
            float u2 = (float)(r1 >> 8) * (1.0f / 16777216.0f);
            nm += sqrtf(-2.0f * __logf(u1)) * __cosf(6.2831853072f * u2);
        }
        nm *= 0.1f;
        ql[(size_t)(b * 64 + h) * NL + blk * NC + c] = mean + nm * (fabsf(sd) + 1e-6f);
    }
}

// ---------------------------------------------------------------------------
// 80x80 fp32 LDS matmul D = alpha * A @ f(B), f fused into B loads:
//   mode 0: B ; 1: 7I-B ; 2: 15I-B ; 3: 13I-B.
// 25 tiles over 8 waves, 20 wmma_f32_16x16x4 per tile.
// ---------------------------------------------------------------------------
__device__ __forceinline__ void mm80(float* __restrict__ D,
                                     const float* __restrict__ A,
                                     const float* __restrict__ B,
                                     int mode, float alpha, int tid) {
    const int wid = tid >> 5, lane = tid & 31;
    const int l15 = lane & 15, hi = lane >> 4;
    const float dconst = (mode == 1) ? 7.0f : (mode == 2) ? 15.0f : 13.0f;
    for (int tile = wid; tile < 25; tile += 8) {
        const int tm = tile / 5, tn = tile % 5;
        const int mrow = tm * 16 + l15;
        const int ncol = tn * 16 + l15;
        v8f acc = {};
        #pragma unroll
        for (int s = 0; s < 20; ++s) {
            const int k0 = 4 * s + 2 * hi;
            v2f a = *(const v2f*)(A + mrow * NP + k0);
            float b0 = B[k0 * NP + ncol];
            float b1 = B[(k0 + 1) * NP + ncol];
            if (mode) {
                b0 = ((k0     == ncol) ? dconst : 0.0f) - b0;
                b1 = ((k0 + 1 == ncol) ? dconst : 0.0f) - b1;
            }
            v2f bv; bv.x = b0; bv.y = b1;
            acc = __builtin_amdgcn_wmma_f32_16x16x4_f32(false, a, false, bv,
                                                        (short)0, acc, false, false);
        }
        const int rbase = tm * 16 + (hi << 3);
        #pragma unroll
        for (int r = 0; r < 8; ++r)
            D[(rbase + r) * NP + tn * 16 + l15] = alpha * acc[r];
    }
}

// ---------------------------------------------------------------------------
// Kernel 3: one workgroup per (b,h). Builds attn2 (identity-padded 80x80),
// Newton pinv (6 iters, 4 WMMA matmuls each), attn3@v and attn1 softmaxes.
// ---------------------------------------------------------------------------
__global__ void __launch_bounds__(256) attn_kernel(const float* __restrict__ qkv,
                                                   const float* __restrict__ ql_g,
                                                   float* __restrict__ out_pre) {
    __shared__ float Smats[5][NP * NP];   // X, plus 4 rotating Z/P/R/T slots
    __shared__ float ql_s[NP], z3_s[NP], w_s[NP], red[NP];
    __shared__ float sMaxr;

    const int tid  = threadIdx.x;
    const int wid  = tid >> 5;
    const int lane = tid & 31;
    const int bh   = blockIdx.x;
    const int b    = bh >> 6;
    const int h    = bh & 63;
    const float* qkvb = qkv + (size_t)b * SEQ * 192;

    // --- init / load landmark row --------------------------------------
    if (tid < NP) {
        ql_s[tid] = (tid < NL) ? ql_g[(size_t)bh * NL + tid] : 0.0f;
        z3_s[tid] = 0.0f;
        w_s[tid]  = 0.0f;
    }
    __syncthreads();

    // --- attn2 = rowsoftmax(ql ql^T), identity-padded into X = Smats[0] --
    float* X = Smats[0];
    for (int j = wid; j < NP; j += 8) {
        const int c0 = lane, c1 = lane + 32, c2 = lane + 64;
        if (j < NL) {
            const float a = ql_s[j];
            const float s0 = (c0 < NL) ? a * ql_s[c0] : -3.0e38f;
            const float s1 = (c1 < NL) ? a * ql_s[c1] : -3.0e38f;
            const float s2 = (c2 < NL) ? a * ql_s[c2] : -3.0e38f;
            const float mx = wave_max_f(fmaxf(s0, fmaxf(s1, s2)));
            const float e0 = (c0 < NL) ? __expf(s0 - mx) : 0.0f;
            const float e1 = (c1 < NL) ? __expf(s1 - mx) : 0.0f;
            const float e2 = (c2 < NL) ? __expf(s2 - mx) : 0.0f;
            const float inv = 1.0f / wave_sum_f(e0 + e1 + e2);
            X[j * NP + c0] = e0 * inv;
            X[j * NP + c1] = e1 * inv;
            if (c2 < NP) X[j * NP + c2] = e2 * inv;
        } else {
            X[j * NP + c0] = (c0 == j) ? 1.0f : 0.0f;
            X[j * NP + c1] = (c1 == j) ? 1.0f : 0.0f;
            if (c2 < NP) X[j * NP + c2] = (c2 == j) ? 1.0f : 0.0f;
        }
    }
    __syncthreads();

    // --- z0 scale: max column-sum (row sums are 1 by softmax) -----------
    if (tid < NP) {
        float cs = 0.0f;
        for (int j = 0; j < NL; ++j) cs += X[j * NP + tid];
        red[tid] = cs;
    }
    __syncthreads();
    if (tid < 32) {
        float m = -3.0e38f;
        for (int c = lane; c < NL; c += 32) m = fmaxf(m, red[c]);
        m = wave_max_f(m);
        if (lane == 0) sMaxr = m;
    }
    __syncthreads();

    // --- Z0 = X^T / maxr (identity on the pad block) into Smats[1] ------
    {
        const float invr = 1.0f / sMaxr;
        float* Z = Smats[1];
        for (int idx = tid; idx < NP * NP; idx += 256) {
            const int j = idx / NP, jc = idx % NP;
            Z[idx] = (j < NL && jc < NL) ? X[jc * NP + j] * invr
                                         : ((j == jc) ? 1.0f : 0.0f);
        }
    }

    // --- z3[j] = softmax_i(ql[j]*k_i) . v  (online softmax per wave) ----
    {
        const float* kp = qkvb + 64 + h;
        const float* vp = qkvb + 128 + h;
        __builtin_prefetch(kp + (size_t)lane * 192, 0, 0);
        __builtin_prefetch(vp + (size_t)lane * 192, 0, 0);
        for (int j = wid; j < NL; j += 8) {
            const float a = ql_s[j];
            float m = -3.0e38f, den = 0.0f, num = 0.0f;
            for (int i = lane; i < SEQ; i += 32) {
                const float kv = kp[(size_t)i * 192];
                const float vv = vp[(size_t)i * 192];
                const float s = a * kv;
                if (s > m) {
                    const float sc = __expf(m - s);
                    den = den * sc + 1.0f;
                    num = num * sc + vv;
                    m = s;
                } else {
                    const float e = __expf(s - m);
                    den += e; num += e * vv;
                }
            }
            #pragma unroll
            for (int msk = 16; msk > 0; msk >>= 1) {      // lane merge
                const float m2 = __shfl_xor(m, msk, 32);
                const float d2 = __shfl_xor(den, msk, 32);
                const float n2 = __shfl_xor(num, msk, 32);
                const float M  = fmaxf(m, m2);
                const float s1 = __expf(m - M), s2 = __expf(m2 - M);
                den = den * s1 + d2 * s2;
                num = num * s1 + n2 * s2;
                m = M;
            }
            if (lane == 0) z3_s[j] = num / den;
        }
    }
    __syncthreads();

    // --- Newton-Schulz pinv: 6 iters x 4 WMMA matmuls -------------------
    int zi = 1, f0 = 2, f1 = 3, f2 = 4;
    for (int it = 0; it < 6; ++it) {
        mm80(Smats[f0], Smats[0],  Smats[zi], 0, 1.0f, tid);  __syncthreads(); // P = X@Z
        mm80(Smats[f1], Smats[f0], Smats[f0], 1, 1.0f, tid);  __syncthreads(); // R = P@(7I-P)
        mm80(Smats[f2], Smats[f0], Smats[f1], 2, 1.0f, tid);  __syncthreads(); // T = P@(15I-R)
        mm80(Smats[f0], Smats[zi], Smats[f2], 3, 0.25f, tid); __syncthreads(); // Z'=.25 Z@(13I-T)
        const int nz = f0; f0 = zi; zi = nz;                  // rotate slots
    }

    // --- w = pinv(attn2) @ z3 -------------------------------------------
    if (tid < NL) {
        const float* zr = Smats[zi] + tid * NP;
        float acc = 0.0f;
        for (int jj = 0; jj < NL; ++jj) acc += zr[jj] * z3_s[jj];
        w_s[tid] = acc;
    }
    __syncthreads();

    // --- out_pre[b,i,h] = softmax_j(q_i*ql_j) . w -----------------------
    {
        const float* qp = qkvb + h;
        for (int i = tid; i < SEQ; i += 256) {
            const float qv = qp[(size_t)i * 192] * SCALE;
            float mx = -3.0e38f;
            for (int j = 0; j < NL; ++j) mx = fmaxf(mx, qv * ql_s[j]);
            float den = 0.0f, num = 0.0f;
            for (int j = 0; j < NL; ++j) {
                const float e = __expf(qv * ql_s[j] - mx);
                den += e; num += e * w_s[j];
            }
            out_pre[((size_t)b * SEQ + i) * 64 + h] = num / den;
        }
    }
}

// ---------------------------------------------------------------------------
extern "C" void kernel_launch(void* const* d_in, const int* in_sizes, int n_in,
                              void* d_out, int out_size, void* d_ws, size_t ws_size,
                              hipStream_t stream) {
    const float* x      = (const float*)d_in[0];   // (2,3900,64)
    const int*   clus   = (const int*)  d_in[1];   // (325,)
    const float* w_qkv  = (const float*)d_in[2];   // (192,64)
    const float* w_out  = (const float*)d_in[3];   // (64,64)
    const float* b_out  = (const float*)d_in[4];   // (64,)
    float* out = (float*)d_out;                    // (2,3900,64)

    float* ws   = (float*)d_ws;
    float* qkv  = ws;                              // 7800*192 = 1,497,600 f
    float* ql   = ws + (size_t)ROWS * 192;         // 128*72   =     9,216 f
    float* opre = ql + (size_t)128 * NL;           // 7800*64  =   499,200 f

    // 1) qkv = x @ w_qkv^T   (488*12 tiles, 4 waves/block)
    {
        const int tiles = ((ROWS + 15) / 16) * (192 / 16);
        gemm_xwT_kernel<<<(tiles + 3) / 4, 128, 0, stream>>>(x, w_qkv, nullptr,
                                                             qkv, ROWS, 192);
    }
    // 2) cluster stats -> landmarks
    landmarks_kernel<<<2 * NB, 64, 0, stream>>>(qkv, clus, ql);
    // 3) attention core (pinv via WMMA) per (b,h)
    attn_kernel<<<128, 256, 0, stream>>>(qkv, ql, opre);
    // 4) out = out_pre @ w_out^T + b_out
    {
        const int tiles = ((ROWS + 15) / 16) * (64 / 16);
        gemm_xwT_kernel<<<(tiles + 3) / 4, 128, 0, stream>>>(opre, w_out, b_out,
                                                             out, ROWS, 64);
    }
    (void)in_sizes; (void)n_in; (void)out_size; (void)ws_size;
}